// GConv_6485400617795
// MI455X (gfx1250) — compile-verified
//
#include <hip/hip_runtime.h>
#include <hip/hip_bf16.h>

typedef float v2f __attribute__((ext_vector_type(2)));
typedef float v8f __attribute__((ext_vector_type(8)));

#define DIN 128
#define BN_EPS 1e-5f

// ---------------- degree / dinv ----------------

__global__ void k_fill_deg(float* deg, int n) {
    int i = blockIdx.x * blockDim.x + threadIdx.x;
    if (i < n) deg[i] = 1.0f;  // self loop
}

__global__ void k_deg_edges(const int* __restrict__ dst, float* deg, int e) {
    int i = blockIdx.x * blockDim.x + threadIdx.x;
    if (i < e) atomicAdd(&deg[dst[i]], 1.0f);
}

__global__ void k_deg_to_dinv(float* dinv, int n) {
    int i = blockIdx.x * blockDim.x + threadIdx.x;
    if (i < n) dinv[i] = rsqrtf(dinv[i]);  // deg >= 1 always (self loop)
}

// ---- fp32 WMMA GEMM + fused self-loop/bias epilogue ----
// One wave per block. Wave computes 16 rows x (CT*16) cols; a single A-tile
// load (16x4 fp32: lane%16 = row, lane/16 selects K-pair) feeds CT WMMAs.
// Epilogue writes raw H (for edge scatter) and Z = bias + h*dinv[row]^2
// (self-loop term of the sym-normalized aggregation, full overwrite).
template <int NC, int CT>
__global__ void k_gemm_wmma_fused(const float* __restrict__ A,
                                  const float* __restrict__ W,
                                  const float* __restrict__ bias,
                                  const float* __restrict__ dinv,
                                  float* __restrict__ H,
                                  float* __restrict__ Z, int K) {
    const int row0 = blockIdx.x * 16;
    const int col0 = blockIdx.y * (CT * 16);
    const int lane = threadIdx.x;
    const int half = lane >> 4;   // 0 or 1
    const int lr   = lane & 15;

    v8f c[CT];
#pragma unroll
    for (int t = 0; t < CT; ++t) c[t] = (v8f){};

    for (int k = 0; k < K; k += 4) {
        const int kh = k + half * 2;
        v2f a;
        a.x = A[(size_t)(row0 + lr) * K + kh];
        a.y = A[(size_t)(row0 + lr) * K + kh + 1];
#pragma unroll
        for (int t = 0; t < CT; ++t) {
            v2f b;
            b.x = W[(size_t)kh * NC + col0 + t * 16 + lr];
            b.y = W[(size_t)(kh + 1) * NC + col0 + t * 16 + lr];
            c[t] = __builtin_amdgcn_wmma_f32_16x16x4_f32(
                       /*neg_a=*/false, a, /*neg_b=*/false, b,
                       /*c_mod=*/(short)0, c[t], /*reuse_a=*/false, /*reuse_b=*/false);
        }
    }

#pragma unroll
    for (int t = 0; t < CT; ++t) {
#pragma unroll
        for (int i = 0; i < 8; ++i) {
            const int row = row0 + half * 8 + i;
            const int col = col0 + t * 16 + lr;
            const float h = c[t][i];
            H[(size_t)row * NC + col] = h;
            const float w = dinv[row];
            Z[(size_t)row * NC + col] = bias[col] + h * w * w;
        }
    }
}

// ---------------- edge aggregation (scatter-add, resolves in L2) ----------------

// out[dst,f] += h[src,f] * dinv[src]*dinv[dst], 4 features per thread
template <int F>
__global__ void k_edge_agg(const float* __restrict__ h, const int* __restrict__ src,
                           const int* __restrict__ dst, const float* __restrict__ dinv,
                           float* __restrict__ out, int e) {
    const int C = F / 4;
    int gid = blockIdx.x * blockDim.x + threadIdx.x;
    if (gid >= e * C) return;
    int ed = gid / C, fc = (gid % C) * 4;
    int s = src[ed], d = dst[ed];
    float w = dinv[s] * dinv[d];
    const float4 hv = *(const float4*)&h[(size_t)s * F + fc];
    float* o = &out[(size_t)d * F + fc];
    atomicAdd(o + 0, hv.x * w);
    atomicAdd(o + 1, hv.y * w);
    atomicAdd(o + 2, hv.z * w);
    atomicAdd(o + 3, hv.w * w);
}

// ---------------- batch norm + PReLU ----------------

__global__ void k_zero_stats(float* stats) {
    int i = blockIdx.x * blockDim.x + threadIdx.x;
    if (i < 512) stats[i] = 0.0f;
}

// stats[0:128] += col sums, stats[128:256] += col sums of squares
__global__ void k_bn_partial(const float* __restrict__ z, float* stats, int n) {
    const int f = threadIdx.x;               // 128 threads = 128 columns
    const int ROWS = 256;
    int r0 = blockIdx.x * ROWS;
    int r1 = r0 + ROWS; if (r1 > n) r1 = n;
    float s = 0.f, q = 0.f;
    for (int r = r0; r < r1; ++r) {
        float v = z[(size_t)r * DIN + f];
        s += v;
        q += v * v;
    }
    atomicAdd(&stats[f], s);
    atomicAdd(&stats[128 + f], q);
}

// stats[256:384] = mean, stats[384:512] = gamma * rsqrt(var + eps)
__global__ void k_bn_finalize(float* stats, const float* __restrict__ gamma, int n) {
    int f = threadIdx.x;
    if (f >= 128) return;
    float inv_n = 1.0f / (float)n;
    float mean = stats[f] * inv_n;
    float var  = stats[128 + f] * inv_n - mean * mean;
    stats[256 + f] = mean;
    stats[384 + f] = gamma[f] * rsqrtf(var + BN_EPS);
}

__global__ void k_bn_apply(float* z, const float* __restrict__ stats,
                           const float* __restrict__ beta,
                           const float* __restrict__ prelu_a, int n) {
    int gid = blockIdx.x * blockDim.x + threadIdx.x;
    if (gid >= n * DIN) return;
    int f = gid % DIN;
    float v = (z[gid] - stats[256 + f]) * stats[384 + f] + beta[f];
    float a = prelu_a[0];
    z[gid] = (v >= 0.f) ? v : a * v;
}

// ---------------- launch ----------------

extern "C" void kernel_launch(void* const* d_in, const int* in_sizes, int n_in,
                              void* d_out, int out_size, void* d_ws, size_t ws_size,
                              hipStream_t stream) {
    const float* x       = (const float*)d_in[0];
    const int*   ei      = (const int*)d_in[1];
    const float* W1      = (const float*)d_in[2];
    const float* b1      = (const float*)d_in[3];
    const float* W2      = (const float*)d_in[4];
    const float* b2      = (const float*)d_in[5];
    const float* gamma   = (const float*)d_in[6];
    const float* beta    = (const float*)d_in[7];
    const float* prelu_a = (const float*)d_in[8];
    float* out = (float*)d_out;

    const int n = in_sizes[0] / DIN;     // 100000 (multiple of 16)
    const int e = in_sizes[1] / 2;       // 1600000
    const int* src = ei;
    const int* dst = ei + e;

    // workspace layout
    char* ws = (char*)d_ws;
    float* h1    = (float*)ws;                                   // n*128 (reused as h2: n*64)
    float* z     = (float*)(ws + (size_t)n * DIN * 4);           // n*128
    float* dinv  = (float*)(ws + (size_t)n * DIN * 4 * 2);       // n   (holds deg first)
    float* stats = dinv + n;                                     // 512 floats

    const int T = 256;

    // degrees -> dinv (needed by the fused GEMM epilogues)
    k_fill_deg<<<(n + T - 1) / T, T, 0, stream>>>(dinv, n);
    k_deg_edges<<<(e + T - 1) / T, T, 0, stream>>>(dst, dinv, e);
    k_deg_to_dinv<<<(n + T - 1) / T, T, 0, stream>>>(dinv, n);

    // layer 1: h1 = x @ W1 (WMMA fp32, 4 col tiles/wave),
    //          z  = b1 + h1*dinv^2 (self-loop), then edge scatter into z
    k_gemm_wmma_fused<128, 4><<<dim3(n / 16, 2), 32, 0, stream>>>(x, W1, b1, dinv, h1, z, DIN);
    k_edge_agg<128><<<((size_t)e * 32 + T - 1) / T, T, 0, stream>>>(h1, src, dst, dinv, z, e);

    // batch norm (training stats) + PReLU, in place on z
    k_zero_stats<<<2, 256, 0, stream>>>(stats);
    k_bn_partial<<<(n + 255) / 256, 128, 0, stream>>>(z, stats, n);
    k_bn_finalize<<<1, 128, 0, stream>>>(stats, gamma, n);
    k_bn_apply<<<((size_t)n * 128 + T - 1) / T, T, 0, stream>>>(z, stats, beta, prelu_a, n);

    // layer 2: h2 = z @ W2 (reuse h1 buffer), out = b2 + h2*dinv^2, edge scatter into out
    k_gemm_wmma_fused<64, 4><<<dim3(n / 16, 1), 32, 0, stream>>>(z, W2, b2, dinv, h1, out, DIN);
    k_edge_agg<64><<<((size_t)e * 16 + T - 1) / T, T, 0, stream>>>(h1, src, dst, dinv, out, e);
}